// TopKSegsSelection_24404004176329
// MI455X (gfx1250) — compile-verified
//
#include <hip/hip_runtime.h>

// Pure gather: out_patch = patch_feat[b, idx[b,k]], out_audio = audio_feat[b, idx[b,k]]
// 129 MB total traffic, 0 FLOPs -> bandwidth-bound (~5.5us at 23.3 TB/s).
// Strategy: wave32-coalesced b128 loads/stores with non-temporal hints,
// wave-uniform scalar index fetch, 14 independent b128 load/store pairs per
// lane for deep LOADcnt/STOREcnt queues.

typedef __attribute__((ext_vector_type(4))) float v4f;

constexpr int B = 16, T = 64, N = 196, C = 512, TOPK = 10;

constexpr int ROW4          = (N * C) / 4;            // 25088 float4 per patch row
constexpr int BLOCK         = 256;                    // 8 wave32s
constexpr int V4_PER_THREAD = 14;                     // 25088 = 7 * 256 * 14
constexpr int PER_BLOCK     = BLOCK * V4_PER_THREAD;  // 3584
constexpr int XBLOCKS       = ROW4 / PER_BLOCK;       // 7
constexpr int AROW4         = C / 4;                  // 128 float4 per audio row
constexpr int A_TOTAL       = B * TOPK * AROW4;       // 20480 float4 total audio

__global__ __launch_bounds__(BLOCK) void topk_gather_kernel(
    const int* __restrict__ idx,      // (B, 1, TOPK) int32
    const v4f* __restrict__ patch,    // (B, T, N, C) f32 as float4
    const v4f* __restrict__ audio,    // (B, T, C)    f32 as float4
    v4f* __restrict__ out_patch,      // (B, TOPK, N, C)
    v4f* __restrict__ out_audio)      // (B, TOPK, C)
{
    if (blockIdx.y < (unsigned)(B * TOPK)) {
        // ---- patch gather: one (b,k) row per blockIdx.y, chunked over x ----
        const int row = blockIdx.y;           // wave-uniform
        const int b   = row / TOPK;
        const int k   = row - b * TOPK;
        const int t   = idx[b * TOPK + k];    // uniform -> scalar (SMEM) load
        const long src = (long)(b * T + t) * ROW4;
        const long dst = (long)row * ROW4;
        const int base = blockIdx.x * PER_BLOCK + threadIdx.x;
#pragma unroll
        for (int j = 0; j < V4_PER_THREAD; ++j) {
            const int off = base + j * BLOCK;             // stays coalesced
            v4f v = __builtin_nontemporal_load(&patch[src + off]);
            __builtin_nontemporal_store(v, &out_patch[dst + off]);
        }
    } else {
        // ---- audio gather: 640 KB, fused to avoid a second launch ----
        const int tid    = blockIdx.x * BLOCK + threadIdx.x;   // 0..1791
        const int stride = XBLOCKS * BLOCK;                    // 1792
        for (int i = tid; i < A_TOTAL; i += stride) {
            const int row = i >> 7;            // / AROW4 (128)
            const int off = i & (AROW4 - 1);
            const int b   = row / TOPK;
            const int k   = row - b * TOPK;
            const int t   = idx[b * TOPK + k];
            v4f v = __builtin_nontemporal_load(&audio[(long)(b * T + t) * AROW4 + off]);
            __builtin_nontemporal_store(v, &out_audio[(long)row * AROW4 + off]);
        }
    }
}

extern "C" void kernel_launch(void* const* d_in, const int* in_sizes, int n_in,
                              void* d_out, int out_size, void* d_ws, size_t ws_size,
                              hipStream_t stream) {
    const int* idx   = (const int*)d_in[0];   // top_k_index_sort (int32)
    const v4f* patch = (const v4f*)d_in[1];   // patch_feat (f32)
    const v4f* audio = (const v4f*)d_in[2];   // audio_feat (f32)

    v4f* out_patch = (v4f*)d_out;
    v4f* out_audio = out_patch + (size_t)B * TOPK * N * C / 4;  // concat order

    dim3 grid(XBLOCKS, B * TOPK + 1);  // y rows 0..159 = patch, y==160 = audio
    topk_gather_kernel<<<grid, BLOCK, 0, stream>>>(idx, patch, audio,
                                                   out_patch, out_audio);
}